// SGNS_53214644798061
// MI455X (gfx1250) — compile-verified
//
#include <hip/hip_runtime.h>
#include <stdint.h>

// SGNS dot-product gather kernel for MI455X (gfx1250, wave32).
//
// Roofline: 16.8 MB random row-gather @ 23.3 TB/s => ~0.7us floor; 4.2 MFLOP
// is free. Strategy: maximize MLP on the gather (32 async global->LDS row
// copies in flight per wave, scalar-base GVS addressing, single asynccnt
// wait, no workgroup barrier), then reduce on the fp32 WMMA pipe
// (V_WMMA_F32_16X16X4_F32; diag(A x B) = the 16 dot products).

typedef __attribute__((ext_vector_type(2))) float v2f;
typedef __attribute__((ext_vector_type(8))) float v8f;

#define NB_DIMS 128
#define ROW_B   (NB_DIMS * 4)  // 512B per embedding row
#define ROW_F   132   // 128 + 4 DWORD pad -> 528B row stride, conflict-free LDS banks
#define WAVES_PER_BLOCK 2
#define ITEMS_PER_WAVE  16
#define ITEMS_PER_BLOCK (WAVES_PER_BLOCK * ITEMS_PER_WAVE)

__global__ __launch_bounds__(WAVES_PER_BLOCK * 32)
void SGNS_wmma_kernel(const int* __restrict__ vii,   // [B, 2] int32 indices
                      const float* __restrict__ W,   // [NB_VECS, 128] fp32
                      float* __restrict__ out,       // [B] fp32
                      int nItems) {
    __shared__ float lds[WAVES_PER_BLOCK][32][ROW_F];

    const int lane = threadIdx.x & 31;
    const int wave = threadIdx.x >> 5;
    const int i0   = blockIdx.x * ITEMS_PER_BLOCK + wave * ITEMS_PER_WAVE;

    // Lane l holds the embedding-row index for LDS row l:
    //   rows 0..15  = target rows of items i0..i0+15
    //   rows 16..31 = context rows of items i0..i0+15
    const int it = i0 + (lane & 15);
    int myIdx = 0;
    if (it < nItems) myIdx = (lane < 16) ? vii[2 * it] : vii[2 * it + 1];

    // ---- Stage 1: async gather, global -> LDS (ASYNCcnt, GVS addressing). ----
    // Row index is wave-uniform per iteration: readlane -> SGPR, so the
    // 64-bit base math is SALU and the per-lane offset VGPR (16*lane) is
    // loop-invariant. One instruction copies one full 512B row (32 x b128),
    // perfectly coalesced (4 x 128B lines, zero over-fetch). All 32 copies
    // (16KB) are in flight before the single asynccnt wait.
    const uint32_t vOff    = 16u * (uint32_t)lane;
    const uint32_t ldsBase = (uint32_t)(uintptr_t)(&lds[wave][0][0]) + vOff;
    #pragma unroll
    for (int r = 0; r < 32; ++r) {
        const uint32_t rowIdx = (uint32_t)__builtin_amdgcn_readlane(myIdx, r);
        const uint64_t gbase  = (uint64_t)(uintptr_t)W + (uint64_t)rowIdx * ROW_B;
        const uint32_t laddr  = ldsBase + (uint32_t)(r * (ROW_F * 4));
        asm volatile("global_load_async_to_lds_b128 %0, %1, %2"
                     :: "v"(laddr), "v"(vOff), "s"(gbase)
                     : "memory");
    }
    // Wave reads only rows it copied itself -> no workgroup barrier needed,
    // just completion of this wave's async copies.
    asm volatile("s_wait_asynccnt 0" ::: "memory");

    // ---- Stage 2: WMMA reduction in fp32. ----
    // A fragment (16x4 f32): lane m (0..15) VGPRs {K=0,K=1}; lanes 16..31 {K=2,K=3}.
    // B fragment (4x16 f32): column N = lane&15, K = vgpr + 2*(lane>>4).
    // For both, lane needs elements [4*kk + 2*h, +1] of LDS row (lane&15).
    const int m = lane & 15;
    const int h = lane >> 4;
    v8f acc = {};
    #pragma unroll
    for (int kk = 0; kk < 32; ++kk) {
        v2f a = *(const v2f*)(&lds[wave][m][4 * kk + 2 * h]);        // target rows
        v2f b = *(const v2f*)(&lds[wave][16 + m][4 * kk + 2 * h]);   // context rows
        acc = __builtin_amdgcn_wmma_f32_16x16x4_f32(
            /*neg_a=*/false, a, /*neg_b=*/false, b,
            /*c_mod=*/(short)0, acc, /*reuse_a=*/false, /*reuse_b=*/false);
    }

    // ---- Diagonal extraction: D[i][i] = dot(target_i, context_i). ----
    // i<8  -> VGPR i,   lane i      (lanes 0..7)
    // i>=8 -> VGPR i-8, lane i+16   (lanes 24..31)
    const int sel = lane & 7;
    float r = acc[0];
    if (sel == 1) r = acc[1];
    if (sel == 2) r = acc[2];
    if (sel == 3) r = acc[3];
    if (sel == 4) r = acc[4];
    if (sel == 5) r = acc[5];
    if (sel == 6) r = acc[6];
    if (sel == 7) r = acc[7];

    const bool lo = (lane < 8);
    const bool hi = (lane >= 24);
    if (lo || hi) {
        const int item = i0 + (lo ? lane : (lane - 16));
        if (item < nItems) out[item] = r;
    }
}

extern "C" void kernel_launch(void* const* d_in, const int* in_sizes, int n_in,
                              void* d_out, int out_size, void* d_ws, size_t ws_size,
                              hipStream_t stream) {
    const int*   vii = (const int*)d_in[0];    // [B,2] int32 (JAX default x64-off)
    const float* W   = (const float*)d_in[1];  // [NB_VECS,128] fp32
    float*       out = (float*)d_out;          // [B] fp32

    const int nItems = out_size;               // B = 16384
    const int blocks = (nItems + ITEMS_PER_BLOCK - 1) / ITEMS_PER_BLOCK;
    SGNS_wmma_kernel<<<blocks, WAVES_PER_BLOCK * 32, 0, stream>>>(vii, W, out, nItems);
}